// NeRFRenderer_1726576857680
// MI455X (gfx1250) — compile-verified
//
#include <hip/hip_runtime.h>
#include <stdint.h>

#define KS 64
#define WAVES_PER_BLOCK 8
#define BLOCK (32 * WAVES_PER_BLOCK)

typedef float v2f __attribute__((ext_vector_type(2)));

// One wave32 per ray. Each lane owns samples k0=2*lane, k1=2*lane+1.
// rgb tile (768 B/ray) is streamed global->LDS with the gfx1250 async path
// (non-temporal: every input byte is single-use) while the wave computes
// alphas + the transmittance scan. All outputs stored non-temporal.
__global__ __launch_bounds__(BLOCK, 8)
void nerf_composite_kernel(const float* __restrict__ rgbs,
                           const float* __restrict__ sigmas,
                           const float* __restrict__ z_samp,
                           float* __restrict__ out_rgb,    // (NR,3)
                           float* __restrict__ out_depth,  // (NR,)
                           float* __restrict__ out_w,      // (NR,K)
                           int nr)
{
    __shared__ float srgb[WAVES_PER_BLOCK][KS * 3];  // 8 * 768 B = 6 KB

    const int wave = threadIdx.x >> 5;
    const int lane = threadIdx.x & 31;
    const int ray  = blockIdx.x * WAVES_PER_BLOCK + wave;
    if (ray >= nr) return;

    // ---- kick async DMA of this ray's rgb row into LDS (3 x 256 B, NT) ----
    // LDS byte address = low 32 bits of the generic pointer (LDS aperture
    // truncates to addr[31:0]). INST_OFFSET adds to BOTH lds and mem address.
    {
        uint64_t g = (uint64_t)(uintptr_t)(rgbs + (size_t)ray * (KS * 3)) + (uint32_t)(lane * 8);
        uint32_t l = (uint32_t)(uintptr_t)(&srgb[wave][0]) + (uint32_t)(lane * 8);
        asm volatile(
            "global_load_async_to_lds_b64 %0, %1, off th:TH_LOAD_NT\n\t"
            "global_load_async_to_lds_b64 %0, %1, off offset:256 th:TH_LOAD_NT\n\t"
            "global_load_async_to_lds_b64 %0, %1, off offset:512 th:TH_LOAD_NT"
            :: "v"(l), "v"(g) : "memory");
    }

    // ---- overlap: load z/sigma pairs (coalesced NT b64), compute alphas ----
    const size_t pbase = (size_t)ray * KS + (size_t)(lane * 2);
    v2f z2 = __builtin_nontemporal_load((const v2f*)(z_samp + pbase));
    v2f s2 = __builtin_nontemporal_load((const v2f*)(sigmas + pbase));

    float z_next = __shfl_down(z2.x, 1);                 // z[2l+2] from lane+1
    float d0 = z2.y - z2.x;
    float d1 = (lane == 31) ? 0.0f : (z_next - z2.y);    // final delta = 0

    float e0 = expf(-d0 * fmaxf(s2.x, 0.0f));            // 1 - alpha0
    float e1 = expf(-d1 * fmaxf(s2.y, 0.0f));            // 1 - alpha1
    float a0 = 1.0f - e0;
    float a1 = 1.0f - e1;
    float t0 = e0 + 1e-10f;                              // shifted term k0
    float t1 = e1 + 1e-10f;                              // shifted term k1

    // ---- multiplicative Hillis-Steele scan across the wave (5 steps) ----
    float incl = t0 * t1;
    #pragma unroll
    for (int off = 1; off < 32; off <<= 1) {
        float v = __shfl_up(incl, off);
        if (lane >= off) incl *= v;
    }
    float excl = __shfl_up(incl, 1);
    if (lane == 0) excl = 1.0f;                          // T before sample 2*lane

    float w0 = a0 * excl;                                // weights[2l]
    float w1 = a1 * excl * t0;                           // weights[2l+1]

    // weights out: coalesced 8 B/lane, non-temporal (never re-read)
    {
        v2f wv; wv.x = w0; wv.y = w1;
        __builtin_nontemporal_store(wv, (v2f*)(out_w + pbase));
    }

    float dep = w0 * z2.x + w1 * z2.y;

    // ---- consume async rgb tile ----
    asm volatile("s_wait_asynccnt 0x0" ::: "memory");
    const float* rp = &srgb[wave][lane * 6];             // 8B-aligned (24B stride)
    v2f c01 = *(const v2f*)(rp);                         // rgb0.r, rgb0.g
    v2f c23 = *(const v2f*)(rp + 2);                     // rgb0.b, rgb1.r
    v2f c45 = *(const v2f*)(rp + 4);                     // rgb1.g, rgb1.b
    float r = w0 * c01.x + w1 * c23.y;
    float g = w0 * c01.y + w1 * c45.x;
    float b = w0 * c23.x + w1 * c45.y;

    // ---- wave32 butterfly reductions (all lanes get the sum) ----
    #pragma unroll
    for (int off = 16; off >= 1; off >>= 1) {
        r   += __shfl_xor(r, off);
        g   += __shfl_xor(g, off);
        b   += __shfl_xor(b, off);
        dep += __shfl_xor(dep, off);
    }

    if (lane < 3) {
        float v = (lane == 0) ? r : ((lane == 1) ? g : b);
        __builtin_nontemporal_store(v, out_rgb + (size_t)ray * 3 + lane);
    }
    if (lane == 3) {
        __builtin_nontemporal_store(dep, out_depth + ray);
    }
}

extern "C" void kernel_launch(void* const* d_in, const int* in_sizes, int n_in,
                              void* d_out, int out_size, void* d_ws, size_t ws_size,
                              hipStream_t stream) {
    const float* rgbs   = (const float*)d_in[0];  // (NR,K,3)
    const float* sigmas = (const float*)d_in[1];  // (NR,K)
    const float* z_samp = (const float*)d_in[2];  // (NR,K)

    const int nr = in_sizes[1] / KS;

    float* out       = (float*)d_out;
    float* out_rgb   = out;                        // NR*3
    float* out_depth = out + (size_t)nr * 3;       // NR
    float* out_w     = out_depth + nr;             // NR*K

    const int blocks = (nr + WAVES_PER_BLOCK - 1) / WAVES_PER_BLOCK;
    hipLaunchKernelGGL(nerf_composite_kernel, dim3(blocks), dim3(BLOCK), 0, stream,
                       rgbs, sigmas, z_samp, out_rgb, out_depth, out_w, nr);
}